// TopKPool_81003083203034
// MI455X (gfx1250) — compile-verified
//
#include <hip/hip_runtime.h>
#include <math.h>

// ---------------- problem constants (match reference) ----------------
#define BGRAPH   64
#define N_PER    1024
#define KKEEP    512
#define NNODES   (BGRAPH * N_PER)      // 65536
#define NEDGE    2097152
#define F_IN     256
#define F_E      16
#define NKEPT    (BGRAPH * KKEEP)      // 32768

// output layout (flat concat, reference return order)
#define OFF_XOUT  ((size_t)0)                              // NKEPT*F_IN floats
#define OFF_EI    (OFF_XOUT + (size_t)NKEPT * F_IN)        // 2*NEDGE ints
#define OFF_EA    (OFF_EI + (size_t)2 * NEDGE)             // NEDGE*F_E floats
#define OFF_BATCH (OFF_EA + (size_t)NEDGE * F_E)           // NKEPT ints
#define OFF_PERM  (OFF_BATCH + (size_t)NKEPT)              // NKEPT ints
#define OFF_SCORE (OFF_PERM + (size_t)NKEPT)               // NNODES floats
#define OFF_MASK  (OFF_SCORE + (size_t)NNODES)             // NEDGE

typedef __attribute__((ext_vector_type(2)))  float v2f;
typedef __attribute__((ext_vector_type(4)))  float f4;
typedef __attribute__((ext_vector_type(8)))  float v8f;

// ---------------------------------------------------------------
// Kernel 0: node_map = -1 sentinel
// ---------------------------------------------------------------
__global__ void tk_init_map(int* __restrict__ node_map) {
    int i = blockIdx.x * blockDim.x + threadIdx.x;
    if (i < NNODES) node_map[i] = -1;
}

// ---------------------------------------------------------------
// Kernel 1: score = tanh(x @ W^T + b), via V_WMMA_F32_16X16X4_F32.
// One wave32 handles 16 rows of x; 64 chained K=4 WMMAs accumulate the
// full K=256 dot product in fp32 (bit-accurate precision class vs ref).
// A layout (16x4 f32): lanes 0-15 = rows, v0/v1 hold K=0/1; lanes 16-31
// hold K=2/3 for the same rows. B is broadcast: every column n gets W[k],
// so D[m][0] (lanes 0 and 16, VGPRs 0..7) is the dot product.
// ---------------------------------------------------------------
__global__ void tk_score_wmma(const float* __restrict__ x,
                              const float* __restrict__ W,
                              const float* __restrict__ bptr,
                              float* __restrict__ ws_score,
                              float* __restrict__ score_out) {
    __shared__ float sW[F_IN];
    int t = threadIdx.x;
    sW[t] = W[t];                       // blockDim.x == 256 == F_IN
    __syncthreads();

    const int wave = t >> 5;            // 8 waves / block
    const int lane = t & 31;
    const int rowbase = (blockIdx.x * 8 + wave) * 16;
    const int m   = lane & 15;          // row within tile
    const int koff = (lane < 16) ? 0 : 2;
    const float* __restrict__ xrow = x + (size_t)(rowbase + m) * F_IN;

    v8f c = {0.f, 0.f, 0.f, 0.f, 0.f, 0.f, 0.f, 0.f};
    #pragma unroll 4
    for (int step = 0; step < F_IN / 4; ++step) {
        const int k0 = step * 4 + koff;
        v2f a, b;
        a.x = xrow[k0];
        a.y = xrow[k0 + 1];
        b.x = sW[k0];
        b.y = sW[k0 + 1];
        // 8 args: (neg_a, A, neg_b, B, c_mod, C, reuse_a, reuse_b)
        c = __builtin_amdgcn_wmma_f32_16x16x4_f32(
                false, a, false, b, (short)0, c, false, false);
    }

    const float bias = bptr[0];
    // D layout: VGPR r, lanes 0-15 -> M=r, N=lane; lanes 16-31 -> M=r+8.
    // Column N=0 lives in lane 0 (M=0..7) and lane 16 (M=8..15).
    if (lane == 0 || lane == 16) {
        const int mb = (lane == 0) ? 0 : 8;
        #pragma unroll
        for (int r = 0; r < 8; ++r) {
            float s = tanhf(c[r] + bias);
            int row = rowbase + mb + r;
            ws_score[row]  = s;
            score_out[row] = s;
        }
    }
}

// ---------------------------------------------------------------
// Kernel 2: per-graph bitonic top-K. One 1024-thread block per graph.
// Descending by score, ties broken by ascending index (jax top_k order).
// ---------------------------------------------------------------
__global__ __launch_bounds__(1024)
void tk_topk(const float* __restrict__ ws_score,
             int* __restrict__ perm_out,
             int* __restrict__ batch_out,
             int* __restrict__ node_map) {
    __shared__ float s_key[N_PER];
    __shared__ int   s_idx[N_PER];
    const int g = blockIdx.x;
    const int t = threadIdx.x;

    s_key[t] = ws_score[g * N_PER + t];
    s_idx[t] = t;
    __syncthreads();

    for (int k = 2; k <= N_PER; k <<= 1) {
        for (int j = k >> 1; j > 0; j >>= 1) {
            int ixj = t ^ j;
            if (ixj > t) {
                float ka = s_key[t],  kb = s_key[ixj];
                int   ia = s_idx[t],  ib = s_idx[ixj];
                bool aFirst = (ka > kb) || (ka == kb && ia < ib); // desc order
                bool wantDesc = ((t & k) == 0);
                if (wantDesc != aFirst) {       // out of order -> swap
                    s_key[t] = kb; s_key[ixj] = ka;
                    s_idx[t] = ib; s_idx[ixj] = ia;
                }
            }
            __syncthreads();
        }
    }

    if (t < KKEEP) {
        const int gnode = g * N_PER + s_idx[t];
        const int oi    = g * KKEEP + t;
        perm_out[oi]  = gnode;
        batch_out[oi] = g;
        node_map[gnode] = oi;
    }
}

// ---------------------------------------------------------------
// Kernel 3: x_out[r] = x[perm[r]] * score[perm[r]] (float4, NT stores).
// 256 threads = 4 rows x 64 float4 lanes.
// ---------------------------------------------------------------
__global__ void tk_gather(const float* __restrict__ x,
                          const float* __restrict__ ws_score,
                          const int* __restrict__ perm,
                          float* __restrict__ xout) {
    const int r = blockIdx.x * 4 + (threadIdx.x >> 6);
    const int cpos = threadIdx.x & 63;
    const int gnode = perm[r];
    const float s = ws_score[gnode];
    const f4* __restrict__ src = (const f4*)(x + (size_t)gnode * F_IN);
    f4 v = src[cpos];
    f4 o = {v.x * s, v.y * s, v.z * s, v.w * s};
    __builtin_nontemporal_store(o, (f4*)(xout + (size_t)r * F_IN) + cpos);
}

// ---------------------------------------------------------------
// Kernel 4: edge filtering + attr stream. One thread per edge.
// node_map table (256KB) sits comfortably in L2; edge_attr streamed NT.
// ---------------------------------------------------------------
__global__ void tk_edges(const int* __restrict__ edge_index,
                         const float* __restrict__ edge_attr,
                         const int* __restrict__ node_map,
                         int* __restrict__ ei_out,
                         float* __restrict__ ea_out,
                         float* __restrict__ mask_out) {
    const int e = blockIdx.x * blockDim.x + threadIdx.x;
    if (e >= NEDGE) return;
    const int r = edge_index[e];
    const int c = edge_index[NEDGE + e];
    const int mr = node_map[r];
    const int mc = node_map[c];
    const bool keep = (mr >= 0) && (mc >= 0);

    ei_out[e]         = keep ? mr : -1;
    ei_out[NEDGE + e] = keep ? mc : -1;
    mask_out[e] = keep ? 1.0f : 0.0f;

    const f4* __restrict__ src = (const f4*)(edge_attr + (size_t)e * F_E);
    f4* __restrict__ dst = (f4*)(ea_out + (size_t)e * F_E);
    #pragma unroll
    for (int i = 0; i < F_E / 4; ++i) {
        f4 v = __builtin_nontemporal_load(src + i);
        f4 o = keep ? v : (f4){0.f, 0.f, 0.f, 0.f};
        __builtin_nontemporal_store(o, dst + i);
    }
}

// ---------------------------------------------------------------
extern "C" void kernel_launch(void* const* d_in, const int* in_sizes, int n_in,
                              void* d_out, int out_size, void* d_ws, size_t ws_size,
                              hipStream_t stream) {
    const float* x          = (const float*)d_in[0];
    const int*   edge_index = (const int*)  d_in[1];
    const float* edge_attr  = (const float*)d_in[2];
    // d_in[3] = batch (unused: graphs are contiguous equal-size)
    const float* W          = (const float*)d_in[4];
    const float* bptr       = (const float*)d_in[5];

    float* out      = (float*)d_out;
    float* xout     = out + OFF_XOUT;
    int*   ei_out   = (int*)(out + OFF_EI);
    float* ea_out   = out + OFF_EA;
    int*   batch_o  = (int*)(out + OFF_BATCH);
    int*   perm_o   = (int*)(out + OFF_PERM);
    float* score_o  = out + OFF_SCORE;
    float* mask_o   = out + OFF_MASK;

    float* ws_score = (float*)d_ws;                    // NNODES floats
    int*   node_map = (int*)((char*)d_ws + (size_t)NNODES * sizeof(float));

    tk_init_map  <<<NNODES / 256, 256, 0, stream>>>(node_map);
    tk_score_wmma<<<NNODES / (16 * 8), 256, 0, stream>>>(x, W, bptr,
                                                         ws_score, score_o);
    tk_topk      <<<BGRAPH, N_PER, 0, stream>>>(ws_score, perm_o, batch_o,
                                                node_map);
    tk_gather    <<<NKEPT / 4, 256, 0, stream>>>(x, ws_score, perm_o, xout);
    tk_edges     <<<NEDGE / 256, 256, 0, stream>>>(edge_index, edge_attr,
                                                   node_map, ei_out, ea_out,
                                                   mask_o);
}